// HandIntersectionLoss_33715493273846
// MI455X (gfx1250) — compile-verified
//
#include <hip/hip_runtime.h>

typedef __attribute__((ext_vector_type(2))) float v2f;
typedef __attribute__((ext_vector_type(8))) float v8f;

#define V_FULL  6890
#define V_HAND  250
#define V_LOOP  20
#define N_FACES 500
#define NPTS    251     // 250 hand verts + 1 loop-mean vert
#define NPAD    256     // padded to 16 WMMA tiles of 16
#define INV_4PI 0.07957747154594767f

// ---------------------------------------------------------------------------
// Kernel 1: gather hand vertex clouds (padded to 256 rows) + zero the output.
// Pad rows get 1e6 so they can never win the min-distance search.
// ---------------------------------------------------------------------------
__global__ __launch_bounds__(256) void gather_zero_kernel(
    const float* __restrict__ verts,
    const int* __restrict__ hl, const int* __restrict__ hr,
    const int* __restrict__ ll, const int* __restrict__ lr,
    float* __restrict__ vl, float* __restrict__ vr,
    float* __restrict__ out)
{
  int b   = blockIdx.x;
  int tid = threadIdx.x;
  if (tid == 0) out[b] = 0.0f;

  const float* vb = verts + (size_t)b * V_FULL * 3;
  for (int side = 0; side < 2; ++side) {
    const int* hi = side ? hr : hl;
    const int* li = side ? lr : ll;
    float* dst = (side ? vr : vl) + (size_t)b * NPAD * 3;
    int i = tid;
    if (i < V_HAND) {
      int idx = hi[i];
      dst[i*3+0] = vb[idx*3+0];
      dst[i*3+1] = vb[idx*3+1];
      dst[i*3+2] = vb[idx*3+2];
    } else if (i == V_HAND) {
      float sx = 0.f, sy = 0.f, sz = 0.f;
      for (int k = 0; k < V_LOOP; ++k) {
        int idx = li[k];
        sx += vb[idx*3+0]; sy += vb[idx*3+1]; sz += vb[idx*3+2];
      }
      const float inv = 1.0f / (float)V_LOOP;
      dst[i*3+0] = sx*inv; dst[i*3+1] = sy*inv; dst[i*3+2] = sz*inv;
    } else if (i < NPAD) {
      dst[i*3+0] = 1.0e6f; dst[i*3+1] = 1.0e6f; dst[i*3+2] = 1.0e6f;
    }
  }
}

// ---------------------------------------------------------------------------
// Kernel 2: one block per (batch, direction).
//   Phase A: winding number per point (VALU + atan2, one thread per point).
//   Phase B: min point-to-vertex distance via V_WMMA_F32_16X16X4_F32 on
//            D = P * V^T, d2 = |p|^2 + |v|^2 - 2*D, cross-lane min reduce.
//            B-fragments built with register selects (no predicated ds_load)
//            and hoisted out of the m-tile loop.
//   Phase C: sum(inside ? depth : 0) -> atomicAdd into out[batch].
// ---------------------------------------------------------------------------
__global__ __launch_bounds__(256) void penetration_kernel(
    const float* __restrict__ vl, const float* __restrict__ vr,
    const int* __restrict__ faces_l, const int* __restrict__ faces_r,
    float* __restrict__ out)
{
  __shared__ float sP[NPAD*3];
  __shared__ float sV[NPAD*3];
  __shared__ float sP2[NPAD];
  __shared__ float sV2[NPAD];
  __shared__ float sTri[N_FACES*9];
  __shared__ float sDepth[NPAD];
  __shared__ float wsum[8];

  const int bid   = blockIdx.x;
  const int batch = bid >> 1;
  const int dir   = bid & 1;
  const float* Pg = (dir == 0 ? vl : vr) + (size_t)batch * NPAD * 3;
  const float* Vg = (dir == 0 ? vr : vl) + (size_t)batch * NPAD * 3;
  const int* faces = (dir == 0) ? faces_r : faces_l;

  const int tid = threadIdx.x;

  for (int i = tid; i < NPAD*3; i += 256) { sP[i] = Pg[i]; sV[i] = Vg[i]; }
  __syncthreads();

  {
    float x = sP[tid*3+0], y = sP[tid*3+1], z = sP[tid*3+2];
    sP2[tid] = x*x + y*y + z*z;
    x = sV[tid*3+0]; y = sV[tid*3+1]; z = sV[tid*3+2];
    sV2[tid] = x*x + y*y + z*z;
  }
  for (int f = tid; f < N_FACES; f += 256) {
    int i0 = faces[f*3+0], i1 = faces[f*3+1], i2 = faces[f*3+2];
    sTri[f*9+0] = sV[i0*3+0]; sTri[f*9+1] = sV[i0*3+1]; sTri[f*9+2] = sV[i0*3+2];
    sTri[f*9+3] = sV[i1*3+0]; sTri[f*9+4] = sV[i1*3+1]; sTri[f*9+5] = sV[i1*3+2];
    sTri[f*9+6] = sV[i2*3+0]; sTri[f*9+7] = sV[i2*3+1]; sTri[f*9+8] = sV[i2*3+2];
  }
  __syncthreads();

  // ---- Phase A: generalized winding number -------------------------------
  float omega = 0.0f;
  if (tid < NPTS) {
    const float px = sP[tid*3+0], py = sP[tid*3+1], pz = sP[tid*3+2];
    for (int f = 0; f < N_FACES; ++f) {
      const float* t = &sTri[f*9];
      float ax = t[0]-px, ay = t[1]-py, az = t[2]-pz;
      float bx = t[3]-px, by = t[4]-py, bz = t[5]-pz;
      float cx = t[6]-px, cy = t[7]-py, cz = t[8]-pz;
      float la = sqrtf(ax*ax + ay*ay + az*az);
      float lb = sqrtf(bx*bx + by*by + bz*bz);
      float lc = sqrtf(cx*cx + cy*cy + cz*cz);
      float ux = by*cz - bz*cy;
      float uy = bz*cx - bx*cz;
      float uz = bx*cy - by*cx;
      float det = ax*ux + ay*uy + az*uz;
      float ab  = ax*bx + ay*by + az*bz;
      float bc  = bx*cx + by*cy + bz*cz;
      float ca  = cx*ax + cy*ay + cz*az;
      float denom = la*lb*lc + ab*lc + bc*la + ca*lb;
      omega += 2.0f * atan2f(det, denom);
    }
  }
  const bool inside = (omega * INV_4PI) > 0.5f;

  // ---- Phase B: min distance via WMMA f32 16x16x4 ------------------------
  const int lane = tid & 31;
  const int wave = tid >> 5;
  const int half = lane >> 4;   // K-half for A/B fragments
  const int lcol = lane & 15;   // M index (A) / N index (B,C,D)

  // Preload all 16 B (vert) fragments; identical for both m-tiles this wave
  // owns. Unconditional LDS loads + register selects -> no exec-mask churn.
  v2f   bf[16];
  float vn2[16];
  #pragma unroll
  for (int u = 0; u < 16; ++u) {
    const int n = u*16 + lcol;
    float bx = sV[n*3+0], by = sV[n*3+1], bz = sV[n*3+2];
    bf[u].x = half ? bz : bx;    // K=0 / K=2
    bf[u].y = half ? 0.0f : by;  // K=1 / K=3 (pad)
    vn2[u]  = sV2[n];
  }

  for (int t = wave*2; t < wave*2 + 2; ++t) {      // 8 waves x 2 = 16 M-tiles
    const int row = t*16 + lcol;
    float axv = sP[row*3+0], ayv = sP[row*3+1], azv = sP[row*3+2];
    v2f afrag;
    afrag.x = half ? azv : axv;
    afrag.y = half ? 0.0f : ayv;

    float p2r[8];
    #pragma unroll
    for (int r = 0; r < 8; ++r) p2r[r] = sP2[t*16 + half*8 + r];

    v8f runmin;
    #pragma unroll
    for (int r = 0; r < 8; ++r) runmin[r] = 3.0e38f;

    #pragma unroll
    for (int u = 0; u < 16; ++u) {                  // 16 N-tiles of verts
      v8f c = {};
      v8f d = __builtin_amdgcn_wmma_f32_16x16x4_f32(
          /*neg_a=*/false, afrag, /*neg_b=*/false, bf[u],
          /*c_mod=*/(short)0, c, /*reuse_a=*/false, /*reuse_b=*/false);
      #pragma unroll
      for (int r = 0; r < 8; ++r) {
        float d2 = p2r[r] + vn2[u] - 2.0f * d[r];
        runmin[r] = fminf(runmin[r], d2);
      }
    }

    // column-min within each 16-lane half (C/D layout: lane = N, vgpr = M)
    #pragma unroll
    for (int r = 0; r < 8; ++r) {
      float m = runmin[r];
      m = fminf(m, __shfl_xor(m, 1, 32));
      m = fminf(m, __shfl_xor(m, 2, 32));
      m = fminf(m, __shfl_xor(m, 4, 32));
      m = fminf(m, __shfl_xor(m, 8, 32));
      if (lcol == 0) sDepth[t*16 + half*8 + r] = sqrtf(m + 1e-12f);
    }
  }
  __syncthreads();

  // ---- Phase C: masked sum reduction -------------------------------------
  float contrib = (tid < NPTS && inside) ? sDepth[tid] : 0.0f;
  #pragma unroll
  for (int off = 16; off >= 1; off >>= 1) contrib += __shfl_xor(contrib, off, 32);
  if (lane == 0) wsum[wave] = contrib;
  __syncthreads();
  if (tid == 0) {
    float s = 0.0f;
    #pragma unroll
    for (int w = 0; w < 8; ++w) s += wsum[w];
    atomicAdd(&out[batch], s);   // exactly 2 commutative adds per batch
  }
}

// ---------------------------------------------------------------------------
extern "C" void kernel_launch(void* const* d_in, const int* in_sizes, int n_in,
                              void* d_out, int out_size, void* d_ws, size_t ws_size,
                              hipStream_t stream) {
  const float* verts = (const float*)d_in[0];
  const int* hl = (const int*)d_in[1];
  const int* hr = (const int*)d_in[2];
  const int* ll = (const int*)d_in[3];
  const int* lr = (const int*)d_in[4];
  const int* fl = (const int*)d_in[5];
  const int* fr = (const int*)d_in[6];
  float* out = (float*)d_out;

  const int B = in_sizes[0] / (V_FULL * 3);

  float* ws = (float*)d_ws;
  float* vl = ws;                               // B * 256 * 3 floats
  float* vr = ws + (size_t)B * NPAD * 3;        // B * 256 * 3 floats

  gather_zero_kernel<<<B, 256, 0, stream>>>(verts, hl, hr, ll, lr, vl, vr, out);
  penetration_kernel<<<B * 2, 256, 0, stream>>>(vl, vr, fl, fr, out);
}